// Slice_13563506720857
// MI455X (gfx1250) — compile-verified
//
#include <hip/hip_runtime.h>

// CDNA5 / gfx1250 bilateral-slice using V_WMMA_F32_16X16X4_F32.
// out[n,c,y,x] = trilinear-slice of grid by guide, recast per 16-pixel group as
// D(16ch x 16px) = A(16ch x 32corners) * B(32corners x 16px) on the matrix pipe.
// A (grid slab) is constant per 32x32 quadrant and lives in 16 VGPRs; B is the
// dense trilinear weight matrix rebuilt per row on VALU (co-executes with XDL).
// Slab staging uses CDNA5 async global->LDS copies (ASYNCcnt-tracked).

typedef __attribute__((ext_vector_type(2))) float v2f;
typedef __attribute__((ext_vector_type(8))) float v8f;

#define GN 8
#define GC 12
#define GD 8
#define GH 16
#define GW 16
#define IMH 1024
#define IMW 1024
#define SLAB_ELEMS (3 * 3 * GD * GC)  // 864 floats

#if __has_builtin(__builtin_amdgcn_wmma_f32_16x16x4_f32)
#define USE_WMMA 1
#endif

#if __has_builtin(__builtin_amdgcn_global_load_async_to_lds_b32) && \
    __has_builtin(__builtin_amdgcn_s_wait_asynccnt)
#define USE_ASYNC_LDS 1
#endif

__device__ __forceinline__ int iclamp(int v, int lo, int hi) {
  return v < lo ? lo : (v > hi ? hi : v);
}

__global__ __launch_bounds__(256)
void bilateral_slice_kernel(const float* __restrict__ grid,   // (8,12,8,16,16)
                            const float* __restrict__ guide,  // (8,1024,1024)
                            float* __restrict__ out) {        // (8,12,1024,1024)
  // 3x3 grid cells (edge-clamped) x all 8 depths x 12 channels for this tile
  __shared__ float slab[SLAB_ELEMS];  // 3456 B

  const int b   = blockIdx.x;
  const int tx  = b & 15;         // 64-pixel tile x (== grid cell x)
  const int ty  = (b >> 4) & 15;  // tile y
  const int n   = b >> 8;         // image
  const int tid = threadIdx.x;

  // ---- stage slab: slab[((j*3+i)*8+p)*12+c] = grid[n,c,p,cy,cx] ----
  for (int e = tid; e < SLAB_ELEMS; e += 256) {
    int c  = e % GC;
    int t  = e / GC;
    int p  = t % GD;
    int t2 = t / GD;
    int i  = t2 % 3;
    int j  = t2 / 3;
    int cy = iclamp(ty - 1 + j, 0, GH - 1);
    int cx = iclamp(tx - 1 + i, 0, GW - 1);
    const float* gsrc = &grid[(((n * GC + c) * GD + p) * GH + cy) * GW + cx];
#ifdef USE_ASYNC_LDS
    // CDNA5 async copy: memory -> LDS with no VGPR staging (ASYNCcnt-tracked).
    // AS casts via integer round-trip; low 32 bits of a flat LDS address are
    // the LDS offset (aperture model). Builtin wants AS(1)/AS(3) int*.
    __attribute__((address_space(1))) int* gp =
        (__attribute__((address_space(1))) int*)(unsigned long long)(size_t)gsrc;
    __attribute__((address_space(3))) int* lp =
        (__attribute__((address_space(3))) int*)(unsigned)(size_t)&slab[e];
    __builtin_amdgcn_global_load_async_to_lds_b32(gp, lp, 0, 0);
#else
    slab[e] = *gsrc;
#endif
  }
#ifdef USE_ASYNC_LDS
  __builtin_amdgcn_s_wait_asynccnt(0);  // my async loads landed in LDS
#endif
  __syncthreads();

  const int lane = tid & 31;
  const int wave = tid >> 5;
  const int q    = wave >> 1;   // quadrant 0..3 of the 64x64 tile
  const int sy   = q >> 1;      // y half: fy = ty + sy - 1 for all its pixels
  const int sx   = q & 1;       // x half
  const int p16  = wave & 1;    // 16-wide pixel group inside the 32-wide half

  // per-lane pixel x (constant across the row loop)
  const int   xloc = sx * 32 + p16 * 16 + (lane & 15);
  const int   x    = tx * 64 + xloc;
  const float wx1  = ((float)xloc + 0.5f) * (1.0f / 64.0f) + 0.5f - (float)sx;
  const float wx0  = 1.0f - wx1;

#ifdef USE_WMMA
  // ---- A: M=channel (16, 12 used), K=corner j (32) ----
  // j = dy*16 + dx*8 + pz ; A[c][j] = slab[sy+dy][sx+dx][pz][c]
  // f32 16x4 layout per K-chunk: lane&15 = M, (lane>>4)*2 = K phase, 2 VGPRs.
  const int ch = lane & 15;
  const int ks = (lane >> 4) * 2;
  v2f A[8];
#pragma unroll
  for (int kk = 0; kk < 8; ++kk) {
    const int j0 = kk * 4 + ks;
    float a0 = 0.0f, a1 = 0.0f;
    if (ch < GC) {
      const int dy = j0 >> 4, dx = (j0 >> 3) & 1;
      const int pz0 = j0 & 7, pz1 = (j0 + 1) & 7;  // j0%4 in {0,2} -> same dy,dx
      const int cell = ((sy + dy) * 3 + (sx + dx)) * GD;
      a0 = slab[(cell + pz0) * GC + ch];
      a1 = slab[(cell + pz1) * GC + ch];
    }
    A[kk].x = a0;
    A[kk].y = a1;
  }

  // Lane's 16 B entries touch only 4 z-planes: {ks, ks+1, ks+4, ks+5}.
  const float pz0f = (float)ks, pz1f = (float)(ks + 1);
  const float pz4f = (float)(ks + 4), pz5f = (float)(ks + 5);

  for (int r = 0; r < 32; ++r) {
    const int   yloc = sy * 32 + r;
    const int   y    = ty * 64 + yloc;
    const float wy1  = ((float)yloc + 0.5f) * (1.0f / 64.0f) + 0.5f - (float)sy;
    const float wy0  = 1.0f - wy1;

    const float g   = guide[(n * IMH + y) * IMW + x];
    // clamped hat == reference's clipped trilinear z-weights (incl. z0==z1 edge)
    const float gzc = fminf(fmaxf(g * (float)GD - 0.5f, 0.0f), 7.0f);
    const float wzp[2][2] = {
        {fmaxf(0.0f, 1.0f - fabsf(gzc - pz0f)),
         fmaxf(0.0f, 1.0f - fabsf(gzc - pz1f))},
        {fmaxf(0.0f, 1.0f - fabsf(gzc - pz4f)),
         fmaxf(0.0f, 1.0f - fabsf(gzc - pz5f))}};

    const float wyx[4] = {wy0 * wx0, wy0 * wx1, wy1 * wx0, wy1 * wx1};

    // ---- B: K=corner j (32), N=pixel (16): dense trilinear weights ----
    // chunk kk: (dy*2+dx) = kk>>1 ; z-plane pair = kk&1
    v2f B[8];
#pragma unroll
    for (int kk = 0; kk < 8; ++kk) {
      const float wv = wyx[kk >> 1];
      B[kk].x = wv * wzp[kk & 1][0];
      B[kk].y = wv * wzp[kk & 1][1];
    }

    v8f acc = {0.f, 0.f, 0.f, 0.f, 0.f, 0.f, 0.f, 0.f};
#pragma unroll
    for (int kk = 0; kk < 8; ++kk) {
      // D(16ch x 16px) += A(16x4) * B(4x16), f32 in/out
      acc = __builtin_amdgcn_wmma_f32_16x16x4_f32(false, A[kk], false, B[kk],
                                                  (short)0, acc, false, false);
    }

    // D layout: VGPR v -> channel v (lanes 0-15) / v+8 (lanes 16-31); N = pixel.
    // VGPRs 0-3: all 32 lanes valid (ch 0-3 & 8-11) -> unconditional stores.
    // VGPRs 4-7: only lanes 0-15 valid (ch 4-7); lanes 16-31 would be ch 12-15.
    const int chanBase = (lane >> 4) * 8;
#pragma unroll
    for (int v = 0; v < 4; ++v) {
      out[((n * GC + chanBase + v) * IMH + y) * IMW + x] = acc[v];
    }
    if (lane < 16) {
#pragma unroll
      for (int v = 4; v < 8; ++v) {
        out[((n * GC + v) * IMH + y) * IMW + x] = acc[v];
      }
    }
  }
#else
  // ---- fallback: direct VALU trilinear (correct, no WMMA) ----
  (void)wx0; (void)wx1; (void)x; (void)p16;
  for (int pidx = tid; pidx < 64 * 64; pidx += 256) {
    const int xl = pidx & 63, yl = pidx >> 6;
    const int px = tx * 64 + xl, py = ty * 64 + yl;
    const float gxv = ((float)px + 0.5f) * (1.0f / 64.0f);
    const float gyv = ((float)py + 0.5f) * (1.0f / 64.0f);
    const float fxf = floorf(gxv - 0.5f), fyf = floorf(gyv - 0.5f);
    const float twx = gxv - 0.5f - fxf, twy = gyv - 0.5f - fyf;
    const int jx0 = (int)fxf - (tx - 1);  // slab-local, in {0,1}
    const int jy0 = (int)fyf - (ty - 1);

    const float g  = guide[(n * IMH + py) * IMW + px];
    const float gz = g * (float)GD;
    const float fz = floorf(gz - 0.5f);
    const float tz = gz - 0.5f - fz;
    const int   z0 = iclamp((int)fz, 0, GD - 1);
    const int   z1 = iclamp((int)fz + 1, 0, GD - 1);

    for (int c = 0; c < GC; ++c) {
      float s = 0.0f;
      for (int dz = 0; dz < 2; ++dz) {
        const int   zz = dz ? z1 : z0;
        const float wz = dz ? tz : (1.0f - tz);
        for (int dy = 0; dy < 2; ++dy) {
          const float wy = dy ? twy : (1.0f - twy);
          for (int dx = 0; dx < 2; ++dx) {
            const float wxv = dx ? twx : (1.0f - twx);
            s += wz * wy * wxv *
                 slab[(((jy0 + dy) * 3 + (jx0 + dx)) * GD + zz) * GC + c];
          }
        }
      }
      out[((n * GC + c) * IMH + py) * IMW + px] = s;
    }
  }
#endif
}

extern "C" void kernel_launch(void* const* d_in, const int* in_sizes, int n_in,
                              void* d_out, int out_size, void* d_ws, size_t ws_size,
                              hipStream_t stream) {
  (void)in_sizes; (void)n_in; (void)out_size; (void)d_ws; (void)ws_size;
  const float* grid  = (const float*)d_in[0];   // (8,12,8,16,16) f32
  const float* guide = (const float*)d_in[1];   // (8,1,1024,1024) f32
  float*       out   = (float*)d_out;           // (8,12,1024,1024) f32
  dim3 blocks(GN * 16 * 16);                    // one block per 64x64 tile/image
  bilateral_slice_kernel<<<blocks, 256, 0, stream>>>(grid, guide, out);
}